// CrossAttention_46780783788034
// MI455X (gfx1250) — compile-verified
//
#include <hip/hip_runtime.h>
#include <math.h>
#include <stdint.h>

typedef __attribute__((ext_vector_type(2))) float v2f;
typedef __attribute__((ext_vector_type(8))) float v8f;

#define B_  2048
#define T_  81
#define C_  128
#define H_  4
#define HD_ 32

// ---------------------------------------------------------------------------
// fp32 WMMA: D(16x16) = A(16x4) * B(4x16) + C   (wave32, gfx1250)
// A fragment: a[v]  = A[m = lane&15][k = (lane>>4)*2 + v]
// B fragment: b[v]  = B[k = (lane>>4)*2 + v][n = lane&15]
// C/D:        c[i] -> (m = i + (lane>>4)*8, n = lane&15)
// ---------------------------------------------------------------------------
static __device__ __forceinline__ v8f wmma_f32(v2f a, v2f b, v8f c) {
  return __builtin_amdgcn_wmma_f32_16x16x4_f32(false, a, false, b, (short)0, c,
                                               false, false);
}

// Async global -> LDS copy, 16 bytes per lane.  VDST VGPR carries the
// wave-relative LDS byte address (generic LDS pointer truncated to 32 bits).
static __device__ __forceinline__ void async_ld_b128(uint32_t lds_byte,
                                                     const float* g) {
  asm volatile("global_load_async_to_lds_b128 %0, %1, off"
               :
               : "v"(lds_byte), "v"(g)
               : "memory");
}
static __device__ __forceinline__ void wait_async0() {
  asm volatile("s_wait_asynccnt 0" ::: "memory");
}
static __device__ __forceinline__ uint32_t lds_off(const void* p) {
  return (uint32_t)(uintptr_t)p;  // low 32 bits of LDS aperture = LDS offset
}

// One 16x16 output tile of [16 x 128] * [128 x 128] with bias-initialized acc.
// a_lds: 16 rows, stride 132 floats (bank-conflict padding).
__device__ __forceinline__ v8f proj_tile(const float* __restrict__ a_lds,
                                         const float* __restrict__ W,
                                         float bias, int ncol, int lm, int lh) {
  v8f c;
#pragma unroll
  for (int i = 0; i < 8; ++i) c[i] = bias;
#pragma unroll
  for (int kc = 0; kc < 32; ++kc) {
    const int k0 = kc * 4 + lh * 2;
    v2f a = *(const v2f*)(a_lds + lm * 132 + k0);
    v2f b;
    b.x = W[(size_t)k0 * C_ + ncol];
    b.y = W[(size_t)(k0 + 1) * C_ + ncol];
    c = wmma_f32(a, b, c);
  }
  return c;
}

// ---------------------------------------------------------------------------
// Kernel 1: fused Q/K/V projections.  grid = (B*T)/16 blocks, 256 threads.
// Each wave computes one 16-col N-tile for q, k and v (3 x 32 WMMA).
// ---------------------------------------------------------------------------
__global__ void __launch_bounds__(256)
qkv_kernel(const float* __restrict__ x, const float* __restrict__ kvs,
           const float* __restrict__ Wq, const float* __restrict__ bq,
           const float* __restrict__ Wk, const float* __restrict__ bk,
           const float* __restrict__ Wv, const float* __restrict__ bv,
           float* __restrict__ qo, float* __restrict__ ko,
           float* __restrict__ vo) {
  __shared__ float xs[16 * 132];
  __shared__ float cs[16 * 132];

  const int tid = threadIdx.x;
  const size_t r0 = (size_t)blockIdx.x * 16;

  // Pin weights in cache early (broadcast across all 10368 blocks).
  __builtin_prefetch(Wq + tid * 64, 0, 0);
  __builtin_prefetch(Wk + tid * 64, 0, 0);
  __builtin_prefetch(Wv + tid * 64, 0, 0);

  // Async-stage the 16x128 activation tiles (512 b128 chunks per array).
  for (int ch = tid; ch < 512; ch += 256) {
    const int row = ch >> 5, c4 = (ch & 31) * 4;
    const uint32_t loff = (uint32_t)(row * 132 + c4) * 4u;
    const size_t g = (r0 + row) * C_ + c4;
    async_ld_b128(lds_off(xs) + loff, x + g);
    async_ld_b128(lds_off(cs) + loff, kvs + g);
  }
  wait_async0();
  __syncthreads();

  const int lane = tid & 31, wave = tid >> 5;
  const int lm = lane & 15, lh = lane >> 4;
  const int ncol = wave * 16 + lm;

  v8f cq = proj_tile(xs, Wq, bq[ncol], ncol, lm, lh);
  v8f ck = proj_tile(cs, Wk, bk[ncol], ncol, lm, lh);
  v8f cv = proj_tile(cs, Wv, bv[ncol], ncol, lm, lh);

#pragma unroll
  for (int i = 0; i < 8; ++i) {
    const size_t row = r0 + (size_t)(i + lh * 8);
    qo[row * C_ + ncol] = cq[i];
    ko[row * C_ + ncol] = ck[i];
    vo[row * C_ + ncol] = cv[i];
  }
}

// ---------------------------------------------------------------------------
// Kernel 2: attention per (b,h).  grid = B*H blocks, 256 threads.
//   S = q k^T * scale + sudoku_mask  -> att_out (pre-softmax, fp32 exact)
//   P = softmax(S) ; Y = P v -> y_ws  ([B,T,C] layout)
// T=81 padded to 96 for 16x16 tiling; padded S entries forced to 0.
// v is held transposed in LDS so Y-phase B fragments are a single b64 load.
// ---------------------------------------------------------------------------
__global__ void __launch_bounds__(256)
attn_kernel(const float* __restrict__ q, const float* __restrict__ k,
            const float* __restrict__ v, float* __restrict__ att_out,
            float* __restrict__ y_ws) {
  __shared__ float qs[96 * 36];    // [t][hd], stride 36
  __shared__ float ks[96 * 36];    // [t][hd], stride 36
  __shared__ float vT[32 * 100];   // [hd][t], stride 100 (transposed)
  __shared__ float S[96 * 100];    // [qi][kj], stride 100

  const int bh = blockIdx.x;
  const int b = bh >> 2;
  const int h = bh & 3;
  const size_t base = (size_t)b * T_ * C_ + (size_t)h * HD_;

  const int tid = threadIdx.x;

  // Zero padding rows of q/k (t = 81..95) and padding cols of vT (t = 81..95).
  for (int idx = tid; idx < 15 * 32; idx += 256) {
    const int r = 81 + (idx >> 5), c = idx & 31;
    qs[r * 36 + c] = 0.f;
    ks[r * 36 + c] = 0.f;
    vT[c * 100 + r] = 0.f;
  }
  // Async-stage valid q/k rows: 81 rows x 8 b128 chunks each.
  for (int ch = tid; ch < 648; ch += 256) {
    const int t = ch >> 3, c4 = (ch & 7) * 4;
    const uint32_t loff = (uint32_t)(t * 36 + c4) * 4u;
    const size_t g = base + (size_t)t * C_ + c4;
    async_ld_b128(lds_off(qs) + loff, q + g);
    async_ld_b128(lds_off(ks) + loff, k + g);
  }
  // Stage v transposed (needs a VGPR round-trip for the transpose).
  for (int idx = tid; idx < 81 * 32; idx += 256) {
    const int t = idx >> 5, hd = idx & 31;
    vT[hd * 100 + t] = v[base + (size_t)t * C_ + hd];
  }
  wait_async0();
  __syncthreads();

  const int lane = tid & 31, wave = tid >> 5;
  const int lm = lane & 15, lh = lane >> 4;
  const float scale = 0.17677669529663687f;  // 1/sqrt(32)

  // --- S = q k^T * scale + mask ; 6x6 tiles of 16x16, K = 32 (8 chunks) ---
  for (int tt = wave; tt < 36; tt += 8) {
    const int mt = tt / 6, nt = tt % 6;
    v8f acc;
#pragma unroll
    for (int i = 0; i < 8; ++i) acc[i] = 0.f;
#pragma unroll
    for (int kc = 0; kc < 8; ++kc) {
      const int k0 = kc * 4 + lh * 2;
      v2f a = *(const v2f*)(qs + (mt * 16 + lm) * 36 + k0);
      v2f bb = *(const v2f*)(ks + (nt * 16 + lm) * 36 + k0);
      acc = wmma_f32(a, bb, acc);
    }
    const int kj = nt * 16 + lm;
    const int rj = kj / 9, cj = kj % 9;
#pragma unroll
    for (int i = 0; i < 8; ++i) {
      const int ml = i + lh * 8;
      const int qi = mt * 16 + ml;
      float sval = 0.0f;  // padded region of S must be exactly zero
      if (qi < T_ && kj < T_) {
        const int ri = qi / 9, ci = qi % 9;
        const bool same = (ri == rj) | (ci == cj) |
                          ((ci / 3 == cj / 3) & (ri / 3 == rj / 3));
        sval = acc[i] * scale + (same ? 0.0f : -INFINITY);
        att_out[((size_t)bh * T_ + qi) * T_ + kj] = sval;
      }
      S[(mt * 16 + ml) * 100 + kj] = sval;
    }
  }
  __syncthreads();

  // --- row softmax over the valid 81 columns (threads 0..80) ---
  if (tid < T_) {
    float mx = -INFINITY;
    for (int j = 0; j < T_; ++j) mx = fmaxf(mx, S[tid * 100 + j]);
    float sum = 0.f;
    for (int j = 0; j < T_; ++j) {
      const float e = __expf(S[tid * 100 + j] - mx);  // exp(-inf)=0 for masked
      S[tid * 100 + j] = e;
      sum += e;
    }
    const float inv = 1.0f / sum;
    for (int j = 0; j < T_; ++j) S[tid * 100 + j] *= inv;
  }
  __syncthreads();

  // --- Y = P v ; 6x2 tiles of 16x16, K = 96 (24 chunks, padding is zero) ---
  for (int tt = wave; tt < 12; tt += 8) {
    const int mt = tt >> 1, nt = tt & 1;
    v8f acc;
#pragma unroll
    for (int i = 0; i < 8; ++i) acc[i] = 0.f;
#pragma unroll
    for (int kc = 0; kc < 24; ++kc) {
      const int k0 = kc * 4 + lh * 2;
      v2f a = *(const v2f*)(S + (mt * 16 + lm) * 100 + k0);
      v2f bb = *(const v2f*)(vT + (nt * 16 + lm) * 100 + k0);
      acc = wmma_f32(a, bb, acc);
    }
#pragma unroll
    for (int i = 0; i < 8; ++i) {
      const int qi = mt * 16 + i + lh * 8;
      if (qi < T_) {
        y_ws[((size_t)b * T_ + qi) * C_ + h * HD_ + nt * 16 + lm] = acc[i];
      }
    }
  }
}

// ---------------------------------------------------------------------------
// Kernel 3: output projection  y = yin * Wp + bp  -> d_out
// ---------------------------------------------------------------------------
__global__ void __launch_bounds__(256)
oproj_kernel(const float* __restrict__ yin, const float* __restrict__ Wp,
             const float* __restrict__ bp, float* __restrict__ out) {
  __shared__ float ys[16 * 132];

  const int tid = threadIdx.x;
  const size_t r0 = (size_t)blockIdx.x * 16;

  __builtin_prefetch(Wp + tid * 64, 0, 0);

  for (int ch = tid; ch < 512; ch += 256) {
    const int row = ch >> 5, c4 = (ch & 31) * 4;
    const uint32_t loff = (uint32_t)(row * 132 + c4) * 4u;
    async_ld_b128(lds_off(ys) + loff, yin + (r0 + row) * C_ + c4);
  }
  wait_async0();
  __syncthreads();

  const int lane = tid & 31, wave = tid >> 5;
  const int lm = lane & 15, lh = lane >> 4;
  const int ncol = wave * 16 + lm;

  v8f c = proj_tile(ys, Wp, bp[ncol], ncol, lm, lh);

#pragma unroll
  for (int i = 0; i < 8; ++i) {
    const size_t row = r0 + (size_t)(i + lh * 8);
    out[row * C_ + ncol] = c[i];
  }
}

// ---------------------------------------------------------------------------
extern "C" void kernel_launch(void* const* d_in, const int* in_sizes, int n_in,
                              void* d_out, int out_size, void* d_ws,
                              size_t ws_size, hipStream_t stream) {
  (void)in_sizes; (void)n_in; (void)out_size; (void)ws_size;

  const float* x  = (const float*)d_in[0];
  const float* kv = (const float*)d_in[1];
  const float* Wq = (const float*)d_in[2];
  const float* bq = (const float*)d_in[3];
  const float* Wk = (const float*)d_in[4];
  const float* bk = (const float*)d_in[5];
  const float* Wv = (const float*)d_in[6];
  const float* bv = (const float*)d_in[7];
  const float* Wp = (const float*)d_in[8];
  const float* bp = (const float*)d_in[9];

  const size_t NS = (size_t)B_ * T_ * C_;   // 21,233,664 floats
  float* y_out   = (float*)d_out;           // [B,T,C]
  float* att_out = y_out + NS;              // [B,H,T,T]

  float* qw = (float*)d_ws;                 // [B,T,C] each
  float* kw = qw + NS;
  float* vw = kw + NS;
  float* yw = vw + NS;

  const int mtiles = (B_ * T_) / 16;        // 10368
  dim3 blk(256);

  qkv_kernel<<<mtiles, blk, 0, stream>>>(x, kv, Wq, bq, Wk, bk, Wv, bv,
                                         qw, kw, vw);
  attn_kernel<<<B_ * H_, blk, 0, stream>>>(qw, kw, vw, att_out, yw);
  oproj_kernel<<<mtiles, blk, 0, stream>>>(yw, Wp, bp, y_out);
}